// WeightedMetaPath2Vec_11020886081827
// MI455X (gfx1250) — compile-verified
//
#include <hip/hip_runtime.h>
#include <stdint.h>

// ---------------- problem constants (from reference) ----------------
#define NN        500000
#define DEG       32
#define EDGES     (NN * DEG)          // 16,000,000
#define DD        128
#define BB        4096
#define WLEN      20                  // WALK_LEN
#define WNODES    (WLEN + 1)          // 21 nodes per walk
#define DUMMY_N   (2 * NN)            // 1,000,000
#define NPAIRS    278528.0f           // 17 * 4096 * 4 per class

typedef __attribute__((ext_vector_type(2))) float v2f;
typedef __attribute__((ext_vector_type(8))) float v8f;

// ---------------- threefry2x32-20 (JAX-compatible) ----------------
__device__ __forceinline__ uint32_t rotl32(uint32_t v, int r) {
  return (v << r) | (v >> (32 - r));
}

__device__ __forceinline__ void threefry2x32(uint32_t k0, uint32_t k1,
                                             uint32_t x0, uint32_t x1,
                                             uint32_t& o0, uint32_t& o1) {
  const uint32_t ks2 = k0 ^ k1 ^ 0x1BD11BDAu;
  uint32_t X0 = x0 + k0, X1 = x1 + k1;
  const int RA[4] = {13, 15, 26, 6};
  const int RB[4] = {17, 29, 16, 24};
#pragma unroll
  for (int r = 0; r < 4; ++r) { X0 += X1; X1 = rotl32(X1, RA[r]); X1 ^= X0; }
  X0 += k1;  X1 += ks2 + 1u;
#pragma unroll
  for (int r = 0; r < 4; ++r) { X0 += X1; X1 = rotl32(X1, RB[r]); X1 ^= X0; }
  X0 += ks2; X1 += k0 + 2u;
#pragma unroll
  for (int r = 0; r < 4; ++r) { X0 += X1; X1 = rotl32(X1, RA[r]); X1 ^= X0; }
  X0 += k0;  X1 += k1 + 3u;
#pragma unroll
  for (int r = 0; r < 4; ++r) { X0 += X1; X1 = rotl32(X1, RB[r]); X1 ^= X0; }
  X0 += k1;  X1 += ks2 + 4u;
#pragma unroll
  for (int r = 0; r < 4; ++r) { X0 += X1; X1 = rotl32(X1, RA[r]); X1 ^= X0; }
  X0 += ks2; X1 += k0 + 5u;
  o0 = X0; o1 = X1;
}

// random_bits element for a 1-D draw of even length `n` under key (k0,k1):
// JAX splits iota(n) into halves (x0=i, x1=i+n/2) and emits (out0, out1).
__device__ __forceinline__ uint32_t jax_bits(uint32_t k0, uint32_t k1,
                                             uint32_t idx, uint32_t half) {
  uint32_t b0, b1;
  if (idx < half) { threefry2x32(k0, k1, idx, idx + half, b0, b1); return b0; }
  threefry2x32(k0, k1, idx - half, idx, b0, b1); return b1;
}

__device__ __forceinline__ float jax_uniform01(uint32_t bits) {
  return __uint_as_float(0x3f800000u | (bits >> 9)) - 1.0f;
}

// JAX _randint for uint32 bits, minval=0, span=NN (16/16 split, u32 wrap)
__device__ __forceinline__ int jax_randint_N(uint32_t bits) {
  const uint32_t span = (uint32_t)NN;
  const uint32_t m1   = 65536u % span;
  const uint32_t mult = (m1 * m1) % span;      // wraps: 65536^2 = 2^32 -> 0 in u32
  uint32_t off = ((bits >> 16) * mult + (bits & 0xFFFFu)) % span;
  return (int)off;
}

// ---------------- kernel 0: zero accumulators ----------------
__global__ void mp2v_init_kernel(float* acc) {
  if (threadIdx.x < 2) acc[threadIdx.x] = 0.0f;
}

// ---------------- kernel 1: positive random walks ----------------
__global__ void mp2v_pos_walk_kernel(const int* __restrict__ batch,
                                     const int* __restrict__ rowptrs,
                                     const int* __restrict__ cols,
                                     const float* __restrict__ prefix,
                                     const int* __restrict__ offsets,
                                     int* __restrict__ walks) {
  int w = blockIdx.x * blockDim.x + threadIdx.x;
  if (w >= BB) return;

  int cur = batch[w];
  int r0 = cur + offsets[0];
  if (r0 > DUMMY_N) r0 = DUMMY_N;
  walks[(size_t)w * WNODES] = r0;

  for (int i = 0; i < WLEN; ++i) {
    const int g = i & 1;
    // k = fold_in(key(1), i) = threefry((0,1), (0,i))
    uint32_t k0, k1;
    threefry2x32(0u, 1u, 0u, (uint32_t)i, k0, k1);
    const uint32_t bits = jax_bits(k0, k1, (uint32_t)w, BB / 2);
    const float u = jax_uniform01(bits);

    const bool maskd = (cur >= DUMMY_N);
    int s = cur;
    if (s < 0) s = 0;
    if (s > NN - 1) s = NN - 1;

    const int* rp = rowptrs + (size_t)g * (NN + 1);
    const int start = rp[s];
    const int end   = rp[s + 1];
    const float* pref = prefix + (size_t)g * EDGES;

    const float prev  = (start > 0) ? pref[start - 1] : 0.0f;
    const float total = pref[end - 1] - prev;
    const float v = u * total + prev;

    // searchsorted(side='right') within [start, end), then clip
    int cnt = 0;
    for (int e = start; e < end; ++e) cnt += (pref[e] <= v) ? 1 : 0;
    int gi = start + cnt;
    if (gi > end - 1) gi = end - 1;
    if (gi < start)   gi = start;
    if (!(total > 0.0f)) gi = 0;

    const int nxt = cols[(size_t)g * EDGES + gi];
    cur = (maskd || !(total > 0.0f)) ? DUMMY_N : nxt;

    int rr = cur + offsets[i + 1];
    if (rr > DUMMY_N) rr = DUMMY_N;
    walks[(size_t)w * WNODES + i + 1] = rr;
  }
}

// ---------------- kernel 2: negative "walks" ----------------
__global__ void mp2v_neg_walk_kernel(const int* __restrict__ batch,
                                     const int* __restrict__ offsets,
                                     int* __restrict__ walks) {
  int w = blockIdx.x * blockDim.x + threadIdx.x;
  if (w >= BB) return;

  walks[(size_t)w * WNODES] = batch[w] + offsets[0];

  const uint32_t half = (uint32_t)(WLEN * BB) / 2u;   // 40960
  for (int i = 0; i < WLEN; ++i) {
    const uint32_t idx = (uint32_t)i * BB + (uint32_t)w; // (WALK_LEN, B) row-major
    const uint32_t bits = jax_bits(0u, 2u, idx, half);   // key(2) = (0,2)
    walks[(size_t)w * WNODES + i + 1] = jax_randint_N(bits) + offsets[i + 1];
  }
}

// ---------------- band extraction (C layout: M = v + 8*(lane>=16), N = lane&15)
__device__ __forceinline__ float band_terms(const v8f& c, int rowbase, int colbase,
                                            int lane, bool isPos) {
  const int m16  = lane & 15;
  const int moff = (lane >> 4) << 3;
  const int n    = colbase + m16;
  float lsum = 0.0f;
#pragma unroll
  for (int v = 0; v < 8; ++v) {
    const int m = rowbase + moff + v;
    const int d = n - m;
    if (d >= 1 && d <= 4 && m <= 16 && n <= 20) {
      const float sig = 1.0f / (1.0f + expf(-c[v]));
      lsum += isPos ? logf(sig + 1e-15f) : logf(1.0f - sig + 1e-15f);
    }
  }
  return lsum;
}

// ---------------- kernel 3: Gram-band loss via f32 WMMA ----------------
// One wave per walk. Stage 21x128 f32 rows in LDS (stride 132: float4-aligned
// and bank-conflict-free for the lane->row pattern). The band of G = X X^T
// needs only tiles (0,0), (0,16), (16,16) of the padded 32x32 Gram matrix,
// and those three tiles share just TWO fragment streams per k-step:
//   P = frag(rows 0..15), Q = frag(rows 16..31)
//   c00 += P*P ; c01 += P*Q ; c11 += Q*Q
// Three independent accumulator chains -> 3-deep WMMA ILP (covers the f32
// XDL latency per the CDNA5 hazard table) with 2 ds_loads per 3 WMMAs.
__global__ void __launch_bounds__(32)
mp2v_loss_kernel(const float* __restrict__ emb,
                 const int* __restrict__ walks,
                 float* __restrict__ acc) {
  __shared__ float X[32][132];   // 16.9 KB

  const int b = blockIdx.x;
  const bool isPos = (b < BB);
  const int* rw = walks + (size_t)b * WNODES;
  const int lane = threadIdx.x;

  // stage embeddings: each lane loads/stores one float4 per row
  for (int r = 0; r < WNODES; ++r) {
    const int node = rw[r];
    const float4 vv = *(const float4*)(emb + (size_t)node * DD + lane * 4);
    *(float4*)&X[r][lane * 4] = vv;
  }
  const float4 z = make_float4(0.f, 0.f, 0.f, 0.f);
  for (int r = WNODES; r < 32; ++r) *(float4*)&X[r][lane * 4] = z;
  __syncthreads();

  // A-fragment build for V_WMMA_F32_16X16X4_F32 (16x4 f32):
  // lane L holds row M = L&15, K-pair k0 + 2*(L>=16). B = X^T builds the same.
  const int m16   = lane & 15;
  const int khalf = (lane >> 4) << 1;            // 0 or 2
  const float* prow = &X[m16][0];                // rows 0..15 stream
  const float* qrow = &X[16 + m16][0];           // rows 16..31 stream

  v8f c00 = {};   // tile (rows 0..15)  x (cols 0..15)
  v8f c01 = {};   // tile (rows 0..15)  x (cols 16..31)
  v8f c11 = {};   // tile (rows 16..31) x (cols 16..31)

#pragma unroll
  for (int k0 = 0; k0 < DD; k0 += 4) {
    v2f p, q;
    p.x = prow[k0 + khalf];  p.y = prow[k0 + khalf + 1];
    q.x = qrow[k0 + khalf];  q.y = qrow[k0 + khalf + 1];
    c00 = __builtin_amdgcn_wmma_f32_16x16x4_f32(false, p, false, p,
                                                (short)0, c00, false, false);
    c01 = __builtin_amdgcn_wmma_f32_16x16x4_f32(false, p, false, q,
                                                (short)0, c01, false, false);
    c11 = __builtin_amdgcn_wmma_f32_16x16x4_f32(false, q, false, q,
                                                (short)0, c11, false, false);
  }

  float lsum = band_terms(c00,  0,  0, lane, isPos)
             + band_terms(c01,  0, 16, lane, isPos)
             + band_terms(c11, 16, 16, lane, isPos);

  // wave32 reduction
  for (int off = 16; off > 0; off >>= 1)
    lsum += __shfl_down(lsum, off, 32);
  if (lane == 0) atomicAdd(&acc[isPos ? 0 : 1], lsum);
}

// ---------------- kernel 4: finalize scalar ----------------
__global__ void mp2v_finalize_kernel(const float* __restrict__ acc,
                                     float* __restrict__ out) {
  if (threadIdx.x == 0) out[0] = -(acc[0] + acc[1]) / NPAIRS;
}

// ---------------- host entry ----------------
extern "C" void kernel_launch(void* const* d_in, const int* in_sizes, int n_in,
                              void* d_out, int out_size, void* d_ws, size_t ws_size,
                              hipStream_t stream) {
  const float* emb     = (const float*)d_in[0];   // (2N+1, 128) f32
  const int*   rowptrs = (const int*)d_in[1];     // (2, N+1)   i32
  const int*   cols    = (const int*)d_in[2];     // (2, E)     i32
  const float* prefix  = (const float*)d_in[3];   // (2, E)     f32
  const int*   batch   = (const int*)d_in[4];     // (B,)       i32
  const int*   offsets = (const int*)d_in[5];     // (21,)      i32
  float* out = (float*)d_out;

  char* ws = (char*)d_ws;
  float* acc       = (float*)ws;                       // 2 f32 accumulators
  int*   pos_walks = (int*)(ws + 256);                 // 4096*21 i32
  int*   neg_walks = pos_walks + (size_t)BB * WNODES;  // 4096*21 i32
  int*   all_walks = pos_walks;                        // contiguous pos|neg

  mp2v_init_kernel<<<1, 32, 0, stream>>>(acc);
  mp2v_pos_walk_kernel<<<(BB + 255) / 256, 256, 0, stream>>>(
      batch, rowptrs, cols, prefix, offsets, pos_walks);
  mp2v_neg_walk_kernel<<<(BB + 255) / 256, 256, 0, stream>>>(
      batch, offsets, neg_walks);
  mp2v_loss_kernel<<<2 * BB, 32, 0, stream>>>(emb, all_walks, acc);
  mp2v_finalize_kernel<<<1, 32, 0, stream>>>(acc, out);
}